// SparseMOE_56160992362635
// MI455X (gfx1250) — compile-verified
//
#include <hip/hip_runtime.h>
#include <hip/hip_bf16.h>

typedef __attribute__((ext_vector_type(2))) float v2f;
typedef __attribute__((ext_vector_type(8))) float v8f;

#define MOE_N     16384
#define MOE_D     512
#define MOE_E     16
#define MOE_K     2
#define M_TILE    64
#define N_TILE    128
#define KC        32
#define MAX_TILES ((MOE_N * MOE_K) / M_TILE + MOE_E)   // 528

// ---------------------------------------------------------------------------
// Kernel 1: router. logits = x @ gate_w + gate_b ; top-2 ; softmax over top-2.
// gate_w (512x16 = 32KB) staged in LDS, one thread per token.
// ---------------------------------------------------------------------------
__global__ __launch_bounds__(256) void moe_router(
    const float* __restrict__ x, const float* __restrict__ gate_w,
    const float* __restrict__ gate_b, int* __restrict__ topk_idx,
    float* __restrict__ topk_w, int* __restrict__ counts)
{
  __shared__ float sGw[MOE_D * MOE_E];   // 32 KB
  __shared__ float sGb[MOE_E];

  int tid = threadIdx.x;
  for (int i = tid; i < MOE_D * MOE_E; i += 256) sGw[i] = gate_w[i];
  if (tid < MOE_E) sGb[tid] = gate_b[tid];
  __syncthreads();

  int n = blockIdx.x * 256 + tid;
  const float* xr = x + (size_t)n * MOE_D;

  float acc[MOE_E];
  #pragma unroll
  for (int e = 0; e < MOE_E; ++e) acc[e] = 0.0f;

  for (int d = 0; d < MOE_D; d += 4) {
    float4 xv = *(const float4*)(xr + d);
    #pragma unroll
    for (int e = 0; e < MOE_E; ++e) {
      acc[e] += xv.x * sGw[(d + 0) * MOE_E + e];
      acc[e] += xv.y * sGw[(d + 1) * MOE_E + e];
      acc[e] += xv.z * sGw[(d + 2) * MOE_E + e];
      acc[e] += xv.w * sGw[(d + 3) * MOE_E + e];
    }
  }

  // top-2, ties resolved to lower index (strict >), matching jax.lax.top_k
  float v0 = -3.402823466e+38f, v1 = -3.402823466e+38f;
  int i0 = 0, i1 = 0;
  #pragma unroll
  for (int e = 0; e < MOE_E; ++e) {
    float v = acc[e] + sGb[e];
    if (v > v0)      { v1 = v0; i1 = i0; v0 = v; i0 = e; }
    else if (v > v1) { v1 = v;  i1 = e; }
  }
  // softmax over {v0, v1}, v0 >= v1
  float ex = __expf(v1 - v0);
  float inv = 1.0f / (1.0f + ex);
  float w0 = inv, w1 = ex * inv;

  topk_idx[2 * n + 0] = i0;  topk_w[2 * n + 0] = w0;
  topk_idx[2 * n + 1] = i1;  topk_w[2 * n + 1] = w1;
  atomicAdd(&counts[i0], 1);
  atomicAdd(&counts[i1], 1);
}

// ---------------------------------------------------------------------------
// Kernel 2: exclusive scan over 16 counts + build M-tile worklist.
// ---------------------------------------------------------------------------
__global__ void moe_scan(const int* __restrict__ counts, int* __restrict__ offsets,
                         int* __restrict__ cursors, int* __restrict__ tileExpert,
                         int* __restrict__ tileRow, int* __restrict__ numTiles)
{
  if (threadIdx.x == 0 && blockIdx.x == 0) {
    int off = 0, nt = 0;
    for (int e = 0; e < MOE_E; ++e) {
      offsets[e] = off;
      cursors[e] = 0;
      int c = counts[e];
      for (int r = 0; r < c; r += M_TILE) { tileExpert[nt] = e; tileRow[nt] = r; ++nt; }
      off += c;
    }
    offsets[MOE_E] = off;
    *numTiles = nt;
  }
}

// ---------------------------------------------------------------------------
// Kernel 3: scatter (token,k) pairs into per-expert segments.
// ---------------------------------------------------------------------------
__global__ __launch_bounds__(256) void moe_scatter(
    const int* __restrict__ topk_idx, const int* __restrict__ offsets,
    int* __restrict__ cursors, int* __restrict__ perm)
{
  int i = blockIdx.x * 256 + threadIdx.x;
  if (i < MOE_N * MOE_K) {
    int e = topk_idx[i];
    int pos = atomicAdd(&cursors[e], 1);
    perm[offsets[e] + pos] = i;
  }
}

// ---------------------------------------------------------------------------
// Kernel 4: grouped GEMM over gathered rows with fp32 WMMA (16x16x4).
// Block = 64x128 output tile, 8 waves in 2x4 layout, each wave owns a 32x32
// register tile (4 accumulators) -> 2x A/B fragment reuse, 4 WMMA per K-step.
// B staged in LDS K-pair-interleaved (Bs2[k>>1][n][k&1]) so every fragment
// (both A and B) is one contiguous 8-byte LDS load -> no operand-assembly movs
// in the WMMA stream.
// Epilogue scatters w * (C + bias) into out via float atomics (2 adds/elem,
// commutative -> deterministic).
// ---------------------------------------------------------------------------
__global__ __launch_bounds__(256) void moe_gemm(
    const float* __restrict__ x, const float* __restrict__ expert_w,
    const float* __restrict__ expert_b, const int* __restrict__ perm,
    const float* __restrict__ topk_w, const int* __restrict__ offsets,
    const int* __restrict__ counts, const int* __restrict__ tileExpert,
    const int* __restrict__ tileRow, const int* __restrict__ numTiles,
    float* __restrict__ out)
{
  int t = blockIdx.x;
  if (t >= *numTiles) return;

  int e        = tileExpert[t];
  int rowBase  = tileRow[t];
  int segStart = offsets[e];
  int rows     = counts[e] - rowBase;
  if (rows > M_TILE) rows = M_TILE;
  int nBase    = blockIdx.y * N_TILE;

  __shared__ float As[M_TILE][KC + 4];        // 64 x 36 ; row stride 144B (8/16B aligned)
  __shared__ float Bs2[KC / 2][N_TILE][2];    // 16 x 128 x 2 = 16KB, K-pair interleaved
  __shared__ int   sTok[M_TILE];
  __shared__ float sW[M_TILE];

  int tid = threadIdx.x;
  if (tid < M_TILE) {
    int tok = 0; float w = 0.0f;
    if (tid < rows) {
      int pi = perm[segStart + rowBase + tid];
      tok = pi >> 1;
      w = topk_w[pi];
    }
    sTok[tid] = tok; sW[tid] = w;
  }
  __syncthreads();

  int lane = tid & 31;
  int wave = tid >> 5;
  int mW = (wave & 1) * 32;    // 2 wave-rows of 32
  int nW = (wave >> 1) * 32;   // 4 wave-cols of 32

  // A staging: rows tid/8 and tid/8+32, 4 floats at col (tid%8)*4  (2 float4/thr)
  int aRow = tid >> 3;                 // 0..31
  int aCol = (tid & 7) * 4;
  const float* aPtr0 = x + (size_t)sTok[aRow]      * MOE_D + aCol;
  const float* aPtr1 = x + (size_t)sTok[aRow + 32] * MOE_D + aCol;
  // B staging: 2x2 (K x N) blocks -> interleaved float4 stores.
  // thread covers k2 in {tid>>6, +4, +8, +12}, n = (tid&63)*2
  int k2t = tid >> 6;                  // 0..3
  int nS  = (tid & 63) * 2;            // 0..126 (even)
  const float* wBase = expert_w + (size_t)e * MOE_D * MOE_D + nBase;

  v8f acc00 = {}, acc01 = {}, acc10 = {}, acc11 = {};

  // per-lane WMMA fragment coordinates (ISA 7.12.2 layouts)
  int m0    = mW + (lane & 15);
  int n0    = nW + (lane & 15);
  int kHalf = (lane >> 4) * 2;        // lanes 0-15: K0/K1 ; lanes 16-31: K2/K3

  for (int k0 = 0; k0 < MOE_D; k0 += KC) {
    float4 a0v = *(const float4*)(aPtr0 + k0);
    float4 a1v = *(const float4*)(aPtr1 + k0);
    float4 bq[4];
    #pragma unroll
    for (int q = 0; q < 4; ++q) {
      int k2 = k2t + q * 4;           // 0..15
      float2 lo = *(const float2*)(wBase + (size_t)(k0 + 2 * k2)     * MOE_D + nS);
      float2 hi = *(const float2*)(wBase + (size_t)(k0 + 2 * k2 + 1) * MOE_D + nS);
      bq[q].x = lo.x; bq[q].y = hi.x; bq[q].z = lo.y; bq[q].w = hi.y;
    }
    __syncthreads();                              // prior compute done reading LDS
    *(float4*)&As[aRow][aCol]      = a0v;
    *(float4*)&As[aRow + 32][aCol] = a1v;
    #pragma unroll
    for (int q = 0; q < 4; ++q)
      *(float4*)&Bs2[k2t + q * 4][nS][0] = bq[q];
    __syncthreads();

    #pragma unroll
    for (int kk = 0; kk < KC; kk += 4) {
      int kf = kk + kHalf;                        // even
      v2f a0 = *(const v2f*)&As[m0][kf];
      v2f a1 = *(const v2f*)&As[m0 + 16][kf];
      v2f b0 = *(const v2f*)&Bs2[kf >> 1][n0][0];
      v2f b1 = *(const v2f*)&Bs2[kf >> 1][n0 + 16][0];
      acc00 = __builtin_amdgcn_wmma_f32_16x16x4_f32(false, a0, false, b0, (short)0, acc00, false, false);
      acc01 = __builtin_amdgcn_wmma_f32_16x16x4_f32(false, a0, false, b1, (short)0, acc01, false, false);
      acc10 = __builtin_amdgcn_wmma_f32_16x16x4_f32(false, a1, false, b0, (short)0, acc10, false, false);
      acc11 = __builtin_amdgcn_wmma_f32_16x16x4_f32(false, a1, false, b1, (short)0, acc11, false, false);
    }
  }

  // Epilogue: C/D layout -> VGPR r holds M=r (lanes 0-15) and M=r+8 (lanes 16-31)
  int mOff = (lane >> 4) * 8;
  float bias0 = expert_b[(size_t)e * MOE_D + nBase + n0];
  float bias1 = expert_b[(size_t)e * MOE_D + nBase + n0 + 16];
  #pragma unroll
  for (int r = 0; r < 8; ++r) {
    int mA = mW + mOff + r;         // tiles acc0x
    int mB = mA + 16;               // tiles acc1x
    if (mA < rows) {
      size_t rowOut = (size_t)sTok[mA] * MOE_D + nBase;
      atomicAdd(&out[rowOut + n0],      sW[mA] * (acc00[r] + bias0));
      atomicAdd(&out[rowOut + n0 + 16], sW[mA] * (acc01[r] + bias1));
    }
    if (mB < rows) {
      size_t rowOut = (size_t)sTok[mB] * MOE_D + nBase;
      atomicAdd(&out[rowOut + n0],      sW[mB] * (acc10[r] + bias0));
      atomicAdd(&out[rowOut + n0 + 16], sW[mB] * (acc11[r] + bias1));
    }
  }
}

// ---------------------------------------------------------------------------
extern "C" void kernel_launch(void* const* d_in, const int* in_sizes, int n_in,
                              void* d_out, int out_size, void* d_ws, size_t ws_size,
                              hipStream_t stream) {
  const float* x        = (const float*)d_in[0];
  const float* gate_w   = (const float*)d_in[1];
  const float* gate_b   = (const float*)d_in[2];
  const float* expert_w = (const float*)d_in[3];
  const float* expert_b = (const float*)d_in[4];
  float* out = (float*)d_out;

  // workspace layout (all 256B aligned)
  char* p = (char*)d_ws;
  auto take = [&](size_t bytes) { char* r = p; p += (bytes + 255) & ~(size_t)255; return r; };
  int*   topk_idx   = (int*)  take(sizeof(int)   * MOE_N * MOE_K);
  float* topk_w     = (float*)take(sizeof(float) * MOE_N * MOE_K);
  int*   counts     = (int*)  take(sizeof(int)   * MOE_E);
  int*   offsets    = (int*)  take(sizeof(int)   * (MOE_E + 1));
  int*   cursors    = (int*)  take(sizeof(int)   * MOE_E);
  int*   perm       = (int*)  take(sizeof(int)   * MOE_N * MOE_K);
  int*   tileExpert = (int*)  take(sizeof(int)   * MAX_TILES);
  int*   tileRow    = (int*)  take(sizeof(int)   * MAX_TILES);
  int*   numTiles   = (int*)  take(sizeof(int));

  hipMemsetAsync(out, 0, sizeof(float) * MOE_N * MOE_D, stream);
  hipMemsetAsync(counts, 0, sizeof(int) * MOE_E, stream);

  moe_router<<<MOE_N / 256, 256, 0, stream>>>(x, gate_w, gate_b, topk_idx, topk_w, counts);
  moe_scan<<<1, 32, 0, stream>>>(counts, offsets, cursors, tileExpert, tileRow, numTiles);
  moe_scatter<<<(MOE_N * MOE_K) / 256, 256, 0, stream>>>(topk_idx, offsets, cursors, perm);

  dim3 grid(MAX_TILES, MOE_D / N_TILE);   // 528 x 4, early-exit on t >= numTiles
  moe_gemm<<<grid, 256, 0, stream>>>(x, expert_w, expert_b, perm, topk_w, offsets,
                                     counts, tileExpert, tileRow, numTiles, out);
}